// Net_68272800137275
// MI455X (gfx1250) — compile-verified
//
#include <hip/hip_runtime.h>
#include <cmath>

// ---------------------------------------------------------------------------
// Fused Canny pipeline for MI455X (gfx1250).
// Memory-bound: 50 MB read + 134 MB written => ~8 us floor @ 23.3 TB/s.
// Single pass, all intermediates in LDS (~32 KB of the 320 KB/WGP).
// gfx1250 paths exercised:
//   * global_load_async_to_lds_b32 + s_wait_asynccnt, software-pipelined
//     across the channel loop (loads for ch+1 overlap compute of ch).
//   * v_wmma_f32_16x16x4_f32: vertical gaussian blur as banded-matrix GEMM,
//     coefficients in A (VALU-built, documented 16x4 layout), data in B
//     (documented column-striped layout), 5 K=4 steps per 16x16 tile.
//     Waves 0-3 do the 32x32 interior as 4 WMMA tiles; waves 4-7 do the
//     width-2 halo ring scalar. Uniform wave branch => EXEC all-ones in WMMA.
// ---------------------------------------------------------------------------

constexpr int TILE = 32;          // output tile (32x32) per 256-thread block
constexpr int NT   = 256;         // 8 waves (wave32)
constexpr int IND  = TILE + 8;    // input tile with halo 4       -> 40
constexpr int INS  = IND + 1;     // LDS row stride (pad)         -> 41
constexpr int HBC  = TILE + 4;    // h-blur cols (halo 2)         -> 36
constexpr int HBS  = HBC + 1;     // 37
constexpr int BLD  = TILE + 4;    // blurred tile (halo 2)        -> 36
constexpr int BLS  = BLD + 1;     // 37
constexpr int GMD  = TILE + 2;    // grad-mag tile (halo 1)       -> 34
constexpr int GMS  = GMD + 1;     // 35

typedef float v2f __attribute__((ext_vector_type(2)));
typedef float v8f __attribute__((ext_vector_type(8)));

#ifndef __HIP_DEVICE_COMPILE__
#define USE_ASYNC_LDS 0
#elif defined(__has_builtin) && __has_builtin(__builtin_amdgcn_global_load_async_to_lds_b32)
#define USE_ASYNC_LDS 1
#else
#define USE_ASYNC_LDS 0
#endif

#if USE_ASYNC_LDS
typedef __attribute__((address_space(1))) int* as1_i32p;
typedef __attribute__((address_space(3))) int* as3_i32p;
#endif

__device__ __forceinline__ void wait_async_zero() {
#if USE_ASYNC_LDS
#if __has_builtin(__builtin_amdgcn_s_wait_asynccnt)
    __builtin_amdgcn_s_wait_asynccnt(0);
#else
    asm volatile("s_wait_asynccnt 0" ::: "memory");
#endif
#endif
}

__global__ __launch_bounds__(NT) void canny_fused_kernel(
    const float* __restrict__ img,     // [3,H,W]
    const float* __restrict__ ghw,     // 5 gaussian taps (unnormalized, std=1)
    float* __restrict__ out_blur,      // [3,H,W]
    float* __restrict__ out_gm,        // [H,W]
    float* __restrict__ out_go,        // [H,W]
    float* __restrict__ out_thin,      // [H,W]
    float* __restrict__ out_thr,       // [H,W]
    float* __restrict__ out_early,     // [H,W]
    int H, int W)
{
    __shared__ float s_in[IND * INS];  // raw input tile (one channel)
    __shared__ float s_hb[IND * HBS];  // horizontally blurred
    __shared__ float s_bl[BLD * BLS];  // fully blurred (zero outside image)
    __shared__ float s_gm[GMD * GMS];  // accumulated gradient magnitude
    __shared__ float s_gx[GMD * GMS];  // accumulated gx
    __shared__ float s_gy[GMD * GMS];  // accumulated gy

    const int tid  = threadIdx.x;
    const int lane = tid & 31;
    const int lm   = lane & 15;        // A: row M  /  B: col N  (documented)
    const int lh   = lane >> 4;        // lane-half selects K-half per VGPR
    const int r0   = blockIdx.y * TILE;
    const int c0   = blockIdx.x * TILE;
    const size_t HW = (size_t)H * (size_t)W;

    // Uniform scalar loads of the gaussian taps (bit-exact vs reference).
    const float g0 = ghw[0], g1 = ghw[1], g2 = ghw[2], g3 = ghw[3], g4 = ghw[4];

    // A-operand: banded vertical-blur matrix M[m,r] = g[r-m], r-m in [0,4],
    // split into five 16x4 K-chunks. A layout: lane holds M=lm, VGPR v holds
    // K = 4k + v + 2*lh.
    v2f Aco[5];
#pragma unroll
    for (int k = 0; k < 5; ++k) {
#pragma unroll
        for (int v = 0; v < 2; ++v) {
            int idx = 4 * k + v + 2 * lh - lm;
            float cf = (idx == 0) ? g0 : (idx == 1) ? g1 : (idx == 2) ? g2
                     : (idx == 3) ? g3 : (idx == 4) ? g4 : 0.0f;
            Aco[k][v] = cf;
        }
    }

    auto issue_loads = [&](int ch) {
        const float* __restrict__ src = img + (size_t)ch * HW;
        for (int idx = tid; idx < IND * IND; idx += NT) {
            int row = idx / IND;
            int col = idx - row * IND;
            int r = r0 - 4 + row;
            int c = c0 - 4 + col;
            float* dst = &s_in[row * INS + col];
            if ((unsigned)r < (unsigned)H && (unsigned)c < (unsigned)W) {
#if USE_ASYNC_LDS
                __builtin_amdgcn_global_load_async_to_lds_b32(
                    (as1_i32p)(src + (size_t)r * W + c),
                    (as3_i32p)dst, 0, 0);
#else
                *dst = src[(size_t)r * W + c];
#endif
            } else {
                *dst = 0.0f;   // zero padding of every conv in the reference
            }
        }
    };

    issue_loads(0);
    wait_async_zero();
    __syncthreads();

    for (int ch = 0; ch < 3; ++ch) {
        // ---- Stage 1: horizontal 5-tap blur: 40 rows x 36 cols ----
        for (int idx = tid; idx < IND * HBC; idx += NT) {
            int row = idx / HBC;
            int j   = idx - row * HBC;
            const float* p = &s_in[row * INS + j];
            s_hb[row * HBS + j] =
                g0 * p[0] + g1 * p[1] + g2 * p[2] + g3 * p[3] + g4 * p[4];
        }
        __syncthreads();               // s_in fully consumed

        // Software pipeline: overlap next channel's global->LDS DMA with the
        // v-blur + sobel compute below.
        if (ch < 2) issue_loads(ch + 1);

        // ---- Stage 2: vertical 5-tap blur (36x36), masked outside image ----
        if (tid < 128) {
            // WMMA path: 32x32 interior = 4 tiles of 16x16, one per wave.
            int w  = tid >> 5;                 // wave 0..3
            int i0 = 2 + ((w >> 1) << 4);      // bl row base: 2 or 18
            int j0 = 2 + ((w & 1) << 4);       // bl col base: 2 or 18
            v8f acc = {};
#pragma unroll
            for (int k = 0; k < 5; ++k) {
                int rbase = i0 + 4 * k + 2 * lh;
                v2f b;                          // B: lane holds col N=lm,
                b.x = s_hb[(rbase + 0) * HBS + j0 + lm];   // VGPR v -> K=v+2lh
                b.y = s_hb[(rbase + 1) * HBS + j0 + lm];
                acc = __builtin_amdgcn_wmma_f32_16x16x4_f32(
                    false, Aco[k], false, b, (short)0, acc, false, false);
            }
            int c = c0 - 2 + j0 + lm;
            bool cin = (unsigned)c < (unsigned)W;
            float* __restrict__ blur_base = out_blur + (size_t)ch * HW;
#pragma unroll
            for (int v = 0; v < 8; ++v) {      // D: row M = v + 8*lh
                int i = i0 + v + 8 * lh;
                int r = r0 - 2 + i;
                bool inimg = cin && ((unsigned)r < (unsigned)H);
                float val = inimg ? acc[v] : 0.0f;
                s_bl[i * BLS + j0 + lm] = val;
                if (inimg) blur_base[(size_t)r * W + c] = acc[v];
            }
        } else {
            // Scalar path: width-2 halo ring of the 36x36 tile (272 values).
            for (int t = tid - 128; t < 272; t += 128) {
                int row, col;
                if (t < 144) {                 // rows 0,1,34,35 full width
                    int band = t / 36;
                    col = t - band * 36;
                    row = (band < 2) ? band : (32 + band);
                } else {                       // cols 0,1,34,35, rows 2..33
                    int u = t - 144;
                    int side = u >> 6;
                    row = 2 + ((u & 63) >> 1);
                    col = (side == 0) ? (u & 1) : (34 + (u & 1));
                }
                const float* p = &s_hb[row * HBS + col];
                float v = g0 * p[0 * HBS] + g1 * p[1 * HBS] + g2 * p[2 * HBS] +
                          g3 * p[3 * HBS] + g4 * p[4 * HBS];
                int r = r0 - 2 + row;
                int c = c0 - 2 + col;
                bool inimg = ((unsigned)r < (unsigned)H) &&
                             ((unsigned)c < (unsigned)W);
                s_bl[row * BLS + col] = inimg ? v : 0.0f;
            }
        }
        __syncthreads();

        // ---- Stage 3: sobel + magnitude, accumulate over channels: 34x34 ----
        for (int idx = tid; idx < GMD * GMD; idx += NT) {
            int i = idx / GMD;
            int j = idx - i * GMD;
            const float* b = &s_bl[i * BLS + j];
            float a00 = b[0],         a01 = b[1],           a02 = b[2];
            float a10 = b[BLS],                             a12 = b[BLS + 2];
            float a20 = b[2 * BLS],   a21 = b[2 * BLS + 1], a22 = b[2 * BLS + 2];
            // cross-correlation with [[1,0,-1],[2,0,-2],[1,0,-1]] and its T
            float gx = (a00 - a02) + 2.0f * (a10 - a12) + (a20 - a22);
            float gy = (a00 + 2.0f * a01 + a02) - (a20 + 2.0f * a21 + a22);
            float m  = sqrtf(gx * gx + gy * gy);
            int r = r0 - 1 + i;
            int c = c0 - 1 + j;
            bool inimg = ((unsigned)r < (unsigned)H) && ((unsigned)c < (unsigned)W);
            if (!inimg) { gx = 0.0f; gy = 0.0f; m = 0.0f; }  // dir-conv zero pad
            int o = i * GMS + j;
            if (ch == 0) { s_gm[o] = m;  s_gx[o] = gx;  s_gy[o] = gy; }
            else         { s_gm[o] += m; s_gx[o] += gx; s_gy[o] += gy; }
        }
        if (ch < 2) wait_async_zero();  // next channel's tile must be resident
        __syncthreads();                // ... and visible to all waves
    }

    // ---- Stage 4: orientation, NMS, thresholds on the 32x32 interior ----
    const float RAD2DEG = (float)(180.0 / 3.14159);   // matches torch source
    for (int q = tid; q < TILE * TILE; q += NT) {
        int ty = q >> 5;
        int tx = q & 31;
        int r = r0 + ty;
        int c = c0 + tx;
        if (r >= H || c >= W) continue;
        int o = (ty + 1) * GMS + (tx + 1);
        float gm  = s_gm[o];
        float sgx = s_gx[o];
        float sgy = s_gy[o];

        float orient = atan2f(sgy, sgx) * RAD2DEG + 180.0f;
        float k  = rintf(orient / 45.0f);   // round-half-even == jnp.round
        float go = k * 45.0f;

        int ik = (int)k;                    // 0..8
        int ip = ik & 7;                    // (k % 8)
        // directional-neighbor LDS delta; offsets for k and k+4 are negatives
        int m3 = ip & 3;
        int dbase = (m3 == 0) ? 1
                  : (m3 == 1) ? (GMS + 1)
                  : (m3 == 2) ? GMS
                  :             (GMS - 1);
        int d = (ip < 4) ? dbase : -dbase;

        float sel_pos = gm - s_gm[o + d];
        float sel_neg = gm - s_gm[o - d];

        float thin_v  = (fminf(sel_pos, sel_neg) > 0.0f) ? gm : 0.0f;
        float thr_v   = (thin_v < 10.0f) ? 0.0f : thin_v;
        float early_v = (gm     < 10.0f) ? 0.0f : gm;

        size_t off = (size_t)r * W + c;
        out_gm[off]    = gm;
        out_go[off]    = go;
        out_thin[off]  = thin_v;
        out_thr[off]   = thr_v;
        out_early[off] = early_v;
    }
}

extern "C" void kernel_launch(void* const* d_in, const int* in_sizes, int n_in,
                              void* d_out, int out_size, void* d_ws, size_t ws_size,
                              hipStream_t stream) {
    const float* img = (const float*)d_in[0];   // [1,3,H,W]
    const float* ghw = (const float*)d_in[1];   // [1,1,1,5] gaussian taps
    (void)n_in; (void)d_ws; (void)ws_size; (void)out_size;

    int HW = in_sizes[0] / 3;
    int H  = (int)std::lround(std::sqrt((double)HW));
    if (H <= 0) H = 1;
    int W  = HW / H;

    float* out   = (float*)d_out;
    float* blur  = out;                       // 3*HW
    float* gm    = out + (size_t)3 * HW;      // HW
    float* go    = out + (size_t)4 * HW;      // HW
    float* thin  = out + (size_t)5 * HW;      // HW
    float* thr   = out + (size_t)6 * HW;      // HW
    float* early = out + (size_t)7 * HW;      // HW

    dim3 grid((W + TILE - 1) / TILE, (H + TILE - 1) / TILE);
    canny_fused_kernel<<<grid, NT, 0, stream>>>(img, ghw, blur, gm, go, thin,
                                                thr, early, H, W);
}